// RectifiedFlowMatcher_81088982548913
// MI455X (gfx1250) — compile-verified
//
#include <hip/hip_runtime.h>
#include <hip/hip_bf16.h>
#include <math.h>

typedef __attribute__((ext_vector_type(2))) float v2f;
typedef __attribute__((ext_vector_type(8))) float v8f;

#define NPTS   2048
#define NBATCH 8
#define TILE   16
#define NTILES (NPTS / TILE)   // 128
#define NWAVES 32              // 1024 threads / wave32
#define LOG2E  1.44269504088896340736f

// raw hardware trans ops (v_exp_f32 / v_log_f32 are natively base-2)
__device__ __forceinline__ float fast_exp2(float x) {
#if __has_builtin(__builtin_amdgcn_exp2f)
  return __builtin_amdgcn_exp2f(x);
#else
  return exp2f(x);
#endif
}
__device__ __forceinline__ float fast_log2(float x) {
#if __has_builtin(__builtin_amdgcn_logf)
  return __builtin_amdgcn_logf(x);
#else
  return log2f(x);
#endif
}

__device__ __forceinline__ float warp_sum(float x) {
#pragma unroll
  for (int off = 16; off > 0; off >>= 1) x += __shfl_xor(x, off, 32);
  return x;
}

// One Sinkhorn half-iteration in base-2 log domain: for each row i of A,
//   dualA[i] = logmass2 - log2( sum_j 2^( dualB[j] - scale2*||A_i - B_j||^2 ) )
// Ak/Bk are the per-lane WMMA operand planes: plane 0 = (x,y), plane 1 = (z,0),
// so the 16x16 Gram tile is one V_WMMA_F32_16X16X4_F32 and the inner loop has
// no cross-half selects. Branchless online logsumexp with a SINGLE exp per
// element: one of the two rescale factors is always exp2(0)=1, so only
// exp2(-|z-mx|) is needed; the rest is cndmask-selected VALU.
__device__ __forceinline__ void sinkhorn_pass(
    const float2* __restrict__ Ak, const float* __restrict__ An,
    const float2* __restrict__ Bk, const float* __restrict__ Bn,
    const float* __restrict__ dualB, float* __restrict__ dualA,
    float scale2, float logmass2, int wave, int half, int l) {
  for (int rb = wave; rb < NTILES; rb += NWAVES) {
    const int row0 = rb * TILE;
    float2 pa = Ak[row0 + l];          // (x,y) or (z,0) per A-matrix layout
    v2f a;
    a.x = pa.x;
    a.y = pa.y;
    float an[8], mx[8], sm[8];
#pragma unroll
    for (int v = 0; v < 8; ++v) {
      an[v] = An[row0 + 8 * half + v]; // |a|^2 for this lane's 8 rows
      mx[v] = -1e30f;
      sm[v] = 0.0f;
    }
    for (int cb = 0; cb < NTILES; ++cb) {
      const int col0 = cb * TILE;
      float2 pb = Bk[col0 + l];
      v2f bv;
      bv.x = pb.x;
      bv.y = pb.y;
      v8f g = {};
      g = __builtin_amdgcn_wmma_f32_16x16x4_f32(false, a, false, bv,
                                                (short)0, g, false, false);
      const float bn = Bn[col0 + l];
      const float db = dualB[col0 + l];
#pragma unroll
      for (int v = 0; v < 8; ++v) {
        float d2 = fmaxf(an[v] + bn - 2.0f * g[v], 0.0f);
        float z = fmaf(-scale2, d2, db);
        float e = fast_exp2(-fabsf(z - mx[v]));   // single trans op
        bool gt = z > mx[v];
        float s_hi = fmaf(sm[v], e, 1.0f);        // z becomes the max
        float s_lo = sm[v] + e;                   // max unchanged
        sm[v] = gt ? s_hi : s_lo;
        mx[v] = gt ? z : mx[v];
      }
    }
    // merge (max,sum) across the 16 lanes of each half (xor masks < 16)
#pragma unroll
    for (int off = 1; off < 16; off <<= 1) {
#pragma unroll
      for (int v = 0; v < 8; ++v) {
        float omx = __shfl_xor(mx[v], off, 32);
        float osm = __shfl_xor(sm[v], off, 32);
        float e = fast_exp2(-fabsf(omx - mx[v]));
        bool gt = omx > mx[v];
        float s_hi = fmaf(sm[v], e, osm);         // other side holds the max
        float s_lo = fmaf(osm, e, sm[v]);         // this side holds the max
        sm[v] = gt ? s_hi : s_lo;
        mx[v] = fmaxf(mx[v], omx);
      }
    }
#pragma unroll
    for (int v = 0; v < 8; ++v)
      if (l == v)
        dualA[row0 + 8 * half + v] = logmass2 - (mx[v] + fast_log2(sm[v]));
  }
}

// mode 0: Sinkhorn assignment (reg=0.01) -> hard argmax -> velocity loss
// mode 1: entropic OT (eps=blur^2=0.0025, C=0.5*d^2) -> sum(P*C)
__global__ __launch_bounds__(1024) void sinkhorn_kernel(
    const float* __restrict__ target, const float* __restrict__ tmpl,
    const float* __restrict__ vpred, float* __restrict__ acc, int mode) {
  __shared__ float2 sAk[2][NPTS];  // plane 0: (x,y), plane 1: (z,0)
  __shared__ float2 sBk[2][NPTS];
  __shared__ float  sAn[NPTS];     // |a|^2
  __shared__ float  sBn[NPTS];     // |b|^2
  __shared__ float  sU[NPTS];      // log2-domain dual u
  __shared__ float  sV[NPTS];      // log2-domain dual v
  const int b = blockIdx.x;
  const int tid = threadIdx.x;

  for (int i = tid; i < NPTS; i += blockDim.x) {
    const int base = (b * NPTS + i) * 3;
    __builtin_prefetch(&vpred[base], 0, 1);
    float x = tmpl[base], y = tmpl[base + 1], z = tmpl[base + 2];
    if (mode == 1) { x += vpred[base]; y += vpred[base + 1]; z += vpred[base + 2]; }
    sAk[0][i] = make_float2(x, y);
    sAk[1][i] = make_float2(z, 0.0f);
    sAn[i] = x * x + y * y + z * z;
    float tx = target[base], ty = target[base + 1], tz = target[base + 2];
    sBk[0][i] = make_float2(tx, ty);
    sBk[1][i] = make_float2(tz, 0.0f);
    sBn[i] = tx * tx + ty * ty + tz * tz;
    sU[i] = 0.0f;
    sV[i] = 0.0f;
  }
  __syncthreads();

  const int wave = tid >> 5, lane = tid & 31;
  const int half = lane >> 4, l = lane & 15;
  const float2* Ak = &sAk[half][0];  // per-lane operand plane, chosen once
  const float2* Bk = &sBk[half][0];
  // base-2 domain: fold log2(e) into the cost scale
  const float scale2 = ((mode == 0) ? 100.0f : 200.0f) * LOG2E;
  const float logmass2 = (mode == 0) ? 0.0f : -11.0f;  // -log2(2048)

  for (int it = 0; it < 50; ++it) {
    sinkhorn_pass(Ak, sAn, Bk, sBn, sV, sU, scale2, logmass2, wave, half, l);
    __syncthreads();
    sinkhorn_pass(Bk, sBn, Ak, sAn, sU, sV, scale2, logmass2, wave, half, l);
    __syncthreads();
  }

  if (mode == 0) {
    // hard_idx[n] = argmax_m ( lv2[m] - scale2*d2(n,m) ); then velocity loss
    float vel = 0.0f;
    for (int rb = wave; rb < NTILES; rb += NWAVES) {
      const int row0 = rb * TILE;
      float2 pa = Ak[row0 + l];
      v2f a;
      a.x = pa.x;
      a.y = pa.y;
      float an[8], bestv[8];
      int besti[8];
#pragma unroll
      for (int v = 0; v < 8; ++v) {
        an[v] = sAn[row0 + 8 * half + v];
        bestv[v] = -1e30f;
        besti[v] = 0;
      }
      for (int cb = 0; cb < NTILES; ++cb) {
        const int col0 = cb * TILE;
        float2 pb = Bk[col0 + l];
        v2f bv;
        bv.x = pb.x;
        bv.y = pb.y;
        v8f g = {};
        g = __builtin_amdgcn_wmma_f32_16x16x4_f32(false, a, false, bv,
                                                  (short)0, g, false, false);
        const float bn = sBn[col0 + l];
        const float lv = sV[col0 + l];
        const int col = col0 + l;
#pragma unroll
        for (int v = 0; v < 8; ++v) {
          float d2 = fmaxf(an[v] + bn - 2.0f * g[v], 0.0f);
          float val = fmaf(-scale2, d2, lv);
          bool gt = val > bestv[v];
          bestv[v] = gt ? val : bestv[v];
          besti[v] = gt ? col : besti[v];
        }
      }
#pragma unroll
      for (int off = 1; off < 16; off <<= 1) {
#pragma unroll
        for (int v = 0; v < 8; ++v) {
          float ov = __shfl_xor(bestv[v], off, 32);
          int oi = __shfl_xor(besti[v], off, 32);
          bool take = ov > bestv[v] || (ov == bestv[v] && oi < besti[v]);
          bestv[v] = take ? ov : bestv[v];
          besti[v] = take ? oi : besti[v];
        }
      }
#pragma unroll
      for (int v = 0; v < 8; ++v) {
        if (l == v) {
          const int row = row0 + 8 * half + v;
          float2 txy = sBk[0][besti[v]];
          float  tz  = sBk[1][besti[v]].x;
          float2 cxy = sAk[0][row];
          float  cz  = sAk[1][row].x;
          const int base = (b * NPTS + row) * 3;
          float dx = vpred[base]     - (txy.x - cxy.x);
          float dy = vpred[base + 1] - (txy.y - cxy.y);
          float dz = vpred[base + 2] - (tz - cz);
          vel += dx * dx + dy * dy + dz * dz;
        }
      }
    }
    vel = warp_sum(vel);
    if (lane == 0) atomicAdd(&acc[0], vel);
  } else {
    // sum(P*C), P = 2^(lu2 + lv2 - scale2*d2), C = 0.5*d2
    float ps = 0.0f;
    for (int rb = wave; rb < NTILES; rb += NWAVES) {
      const int row0 = rb * TILE;
      float2 pa = Ak[row0 + l];
      v2f a;
      a.x = pa.x;
      a.y = pa.y;
      float an[8], lu[8];
#pragma unroll
      for (int v = 0; v < 8; ++v) {
        an[v] = sAn[row0 + 8 * half + v];
        lu[v] = sU[row0 + 8 * half + v];
      }
      for (int cb = 0; cb < NTILES; ++cb) {
        const int col0 = cb * TILE;
        float2 pb = Bk[col0 + l];
        v2f bv;
        bv.x = pb.x;
        bv.y = pb.y;
        v8f g = {};
        g = __builtin_amdgcn_wmma_f32_16x16x4_f32(false, a, false, bv,
                                                  (short)0, g, false, false);
        const float bn = sBn[col0 + l];
        const float lv = sV[col0 + l];
#pragma unroll
        for (int v = 0; v < 8; ++v) {
          float d2 = fmaxf(an[v] + bn - 2.0f * g[v], 0.0f);
          float z = fmaf(-scale2, d2, lu[v] + lv);
          ps += fast_exp2(z) * (0.5f * d2);
        }
      }
    }
    ps = warp_sum(ps);
    if (lane == 0) atomicAdd(&acc[1], ps);
  }
}

__global__ __launch_bounds__(1024) void chamfer_repulsion_kernel(
    const float* __restrict__ target, const float* __restrict__ tmpl,
    const float* __restrict__ vpred, float* __restrict__ acc) {
  __shared__ float4 sX[NPTS];   // x1 = template + v_pred, w = |p|^2
  __shared__ float4 sT[NPTS];   // target, w = |p|^2
  const int b = blockIdx.x;
  const int tid = threadIdx.x;
  for (int i = tid; i < NPTS; i += blockDim.x) {
    const int base = (b * NPTS + i) * 3;
    float x = tmpl[base] + vpred[base];
    float y = tmpl[base + 1] + vpred[base + 1];
    float z = tmpl[base + 2] + vpred[base + 2];
    sX[i] = make_float4(x, y, z, x * x + y * y + z * z);
    float tx = target[base], ty = target[base + 1], tz = target[base + 2];
    sT[i] = make_float4(tx, ty, tz, tx * tx + ty * ty + tz * tz);
  }
  __syncthreads();

  float chamA = 0.0f, chamB = 0.0f, rep = 0.0f;
  for (int r = tid; r < NPTS; r += blockDim.x) {
    float4 p = sX[r];
    float mn = 1e30f;
    for (int m = 0; m < NPTS; ++m) {
      float4 q = sT[m];
      float d = fmaxf(p.w + q.w - 2.0f * (p.x * q.x + p.y * q.y + p.z * q.z), 0.0f);
      mn = fminf(mn, d);
    }
    chamA += mn;
    // repulsion: 9 smallest self-distances (self 0 dropped below)
    float knn[9];
#pragma unroll
    for (int i = 0; i < 9; ++i) knn[i] = 1e30f;
    for (int m = 0; m < NPTS; ++m) {
      float4 q = sX[m];
      float d = fmaxf(p.w + q.w - 2.0f * (p.x * q.x + p.y * q.y + p.z * q.z), 0.0f);
      if (d < knn[8]) {   // rarely taken: EXEC-skip is a win here
        knn[8] = d;
#pragma unroll
        for (int i = 8; i > 0; --i) {
          if (knn[i] < knn[i - 1]) { float t = knn[i]; knn[i] = knn[i - 1]; knn[i - 1] = t; }
        }
      }
    }
#pragma unroll
    for (int i = 1; i < 9; ++i) rep += 1.0f / (knn[i] + 1e-8f);
  }
  for (int r = tid; r < NPTS; r += blockDim.x) {
    float4 p = sT[r];
    float mn = 1e30f;
    for (int m = 0; m < NPTS; ++m) {
      float4 q = sX[m];
      float d = fmaxf(p.w + q.w - 2.0f * (p.x * q.x + p.y * q.y + p.z * q.z), 0.0f);
      mn = fminf(mn, d);
    }
    chamB += mn;
  }
  chamA = warp_sum(chamA);
  chamB = warp_sum(chamB);
  rep   = warp_sum(rep);
  if ((tid & 31) == 0) {
    atomicAdd(&acc[2], chamA);
    atomicAdd(&acc[3], chamB);
    atomicAdd(&acc[4], rep);
  }
}

__global__ void reg_kernel(const float* __restrict__ tmpl, float* __restrict__ acc) {
  float s = 0.0f;
  for (int i = threadIdx.x; i < NPTS; i += blockDim.x) {
    const int base = i * 3;  // batch 0 only
    float x = tmpl[base], y = tmpl[base + 1], z = tmpl[base + 2];
    float d = sqrtf(x * x + y * y + z * z) - 1.5f;
    s += d * d;
  }
  s = warp_sum(s);
  if ((threadIdx.x & 31) == 0) atomicAdd(&acc[5], s);
}

__global__ void zero_kernel(float* __restrict__ acc) {
  if (threadIdx.x < 8) acc[threadIdx.x] = 0.0f;
}

__global__ void finalize_kernel(const float* __restrict__ acc, float* __restrict__ out) {
  const float BN = (float)NBATCH * (float)NPTS;
  float loss = acc[0] / (BN * 3.0f)                    // velocity (mean over B,N,3)
             + 0.1f  * (acc[1] / (float)NBATCH)        // LAMBDA_OT * mean_b sum(P*C)
             + 0.01f * (acc[5] / (float)NPTS)          // LAMBDA_REG * mean
             + 0.1f  * ((acc[2] + acc[3]) / BN)        // LAMBDA_CHAMFER * (mean+mean)
             + 0.01f * (acc[4] / (BN * 8.0f));         // LAMBDA_REPULSION * mean over k
  out[0] = loss;
}

extern "C" void kernel_launch(void* const* d_in, const int* in_sizes, int n_in,
                              void* d_out, int out_size, void* d_ws, size_t ws_size,
                              hipStream_t stream) {
  const float* target = (const float*)d_in[0];
  const float* tmpl   = (const float*)d_in[1];
  const float* vpred  = (const float*)d_in[2];
  // d_in[3] = t : unused by the loss (x_t is dead in the reference)
  float* out = (float*)d_out;
  float* acc = (float*)d_ws;  // 8 floats of accumulators

  zero_kernel<<<1, 32, 0, stream>>>(acc);
  sinkhorn_kernel<<<NBATCH, 1024, 0, stream>>>(target, tmpl, vpred, acc, 0);
  sinkhorn_kernel<<<NBATCH, 1024, 0, stream>>>(target, tmpl, vpred, acc, 1);
  chamfer_repulsion_kernel<<<NBATCH, 1024, 0, stream>>>(target, tmpl, vpred, acc);
  reg_kernel<<<1, 256, 0, stream>>>(tmpl, acc);
  finalize_kernel<<<1, 1, 0, stream>>>(acc, out);
}